// SparseAttentionHead_27779848471219
// MI455X (gfx1250) — compile-verified
//
#include <hip/hip_runtime.h>
#include <math.h>

// ---- CDNA5 WMMA fragment types (f32 16x16x4) ----
typedef __attribute__((ext_vector_type(2))) float v2f;
typedef __attribute__((ext_vector_type(8))) float v8f;

#define S_LEN    4096
#define D_HEAD   128
#define QT       16          // query rows per workgroup
#define K_SELECT 819         // max(1, int(4096 * (1-0.8)))
#define NWAVES   8
#define BLOCK    256
#define S_STRIDE (S_LEN + 4) // LDS row stride (floats), pad to dodge bank conflicts

// Map float -> monotonically ordered uint (larger float => larger key)
__device__ __forceinline__ unsigned fkey(float f) {
  unsigned u = __float_as_uint(f);
  return (u & 0x80000000u) ? ~u : (u | 0x80000000u);
}
__device__ __forceinline__ float funkey(unsigned k) {
  return __uint_as_float((k & 0x80000000u) ? (k ^ 0x80000000u) : ~k);
}

__global__ __launch_bounds__(BLOCK)
void sparse_attn_kernel(const float* __restrict__ Q,
                        const float* __restrict__ Km,
                        const float* __restrict__ V,
                        const float* __restrict__ temp,
                        float* __restrict__ out,
                        float* __restrict__ attn)
{
  extern __shared__ char smem_raw[];
  float*    s_scores = (float*)smem_raw;                           // [QT][S_STRIDE]
  unsigned* s_hist   = (unsigned*)(smem_raw + QT * S_STRIDE * 4);  // [256]
  float*    s_kth    = (float*)(s_hist + 256);                     // [QT]

  const int tid  = threadIdx.x;
  const int lane = tid & 31;
  const int wave = tid >> 5;
  const int q0   = blockIdx.x * QT;    // query tile base
  const int b    = blockIdx.y;         // batch

  const float inv_scale = rsqrtf((float)D_HEAD) / fmaxf(temp[0], 0.1f);

  const float* Qb = Q  + ((size_t)b * S_LEN + q0) * D_HEAD;
  const float* Kb = Km + (size_t)b * S_LEN * D_HEAD;
  const float* Vb = V  + (size_t)b * S_LEN * D_HEAD;

  // Per-lane WMMA addressing: lane holds matrix row m, K-slice base koff.
  const int m      = lane & 15;          // A-row / B-col within 16-tile
  const int koff   = (lane >> 4) * 2;    // 0 or 2 within the K=4 step
  const int row_hi = (lane >> 4) * 8;    // C/D row offset for this half-wave

  // ===================== Phase 1: scores = (Q K^T) * scale =====================
  // Preload the Q fragment for all 32 K-steps; FULL unroll everywhere so qa[]
  // stays in VGPRs (partial unroll previously demoted it to scratch).
  float qa[64];
  #pragma unroll
  for (int s = 0; s < 32; ++s) {
    const int d = s * 4 + koff;
    qa[2 * s]     = Qb[m * D_HEAD + d];
    qa[2 * s + 1] = Qb[m * D_HEAD + d + 1];
  }

  for (int t = wave; t < S_LEN / 16; t += NWAVES) {
    const int k0 = t * 16;
    v8f acc = {};
    const float* kptr = Kb + (size_t)(k0 + m) * D_HEAD + koff;
    #pragma unroll
    for (int s = 0; s < 32; ++s) {
      v2f a;  a.x  = qa[2 * s];     a.y  = qa[2 * s + 1];
      v2f bb; bb.x = kptr[s * 4];   bb.y = kptr[s * 4 + 1];
      acc = __builtin_amdgcn_wmma_f32_16x16x4_f32(
          false, a, false, bb, (short)0, acc, false, false);
    }
    #pragma unroll
    for (int r = 0; r < 8; ++r)
      s_scores[(r + row_hi) * S_STRIDE + k0 + m] = acc[r] * inv_scale;
  }
  __syncthreads();

  // ============ Phase 2: exact kth-largest per row (4-pass radix select) ============
  for (int row = 0; row < QT; ++row) {
    unsigned prefix = 0;
    int remaining = K_SELECT;
    for (int pass = 3; pass >= 0; --pass) {
      const int shift = pass * 8;
      s_hist[tid] = 0;                       // BLOCK == 256 bins
      __syncthreads();
      const unsigned pmask = (pass == 3) ? 0u : (0xFFFFFFFFu << (shift + 8));
      for (int i = tid; i < S_LEN; i += BLOCK) {
        const unsigned u = fkey(s_scores[row * S_STRIDE + i]);
        if ((u & pmask) == (prefix & pmask))
          atomicAdd(&s_hist[(u >> shift) & 0xFFu], 1u);
      }
      __syncthreads();
      if (tid == 0) {
        int rem = remaining;
        unsigned bin = 0;
        for (int bnum = 255; bnum >= 0; --bnum) {
          const int c = (int)s_hist[bnum];
          if (rem <= c) { bin = (unsigned)bnum; break; }
          rem -= c;
        }
        s_hist[0] = bin;
        s_hist[1] = (unsigned)rem;
      }
      __syncthreads();
      prefix   |= (s_hist[0] & 0xFFu) << shift;
      remaining = (int)s_hist[1];
      __syncthreads();                        // protect s_hist before next clear
    }
    if (tid == 0) s_kth[row] = funkey(prefix);
    __syncthreads();
  }

  // ====== Phase 3: thresholded softmax (2 rows per wave, shuffle reductions) ======
  for (int rr = 0; rr < 2; ++rr) {
    const int row  = wave * 2 + rr;
    const float kth = s_kth[row];
    float* srow = &s_scores[row * S_STRIDE];

    float mx = -INFINITY;                     // row max == max of selected set
    for (int i = lane; i < S_LEN; i += 32) mx = fmaxf(mx, srow[i]);
    #pragma unroll
    for (int off = 16; off > 0; off >>= 1) mx = fmaxf(mx, __shfl_xor(mx, off, 32));

    float sum = 0.f;
    for (int i = lane; i < S_LEN; i += 32) {
      const float s = srow[i];
      const float w = (s >= kth) ? __expf(s - mx) : 0.f;
      srow[i] = w;
      sum += w;
    }
    #pragma unroll
    for (int off = 16; off > 0; off >>= 1) sum += __shfl_xor(sum, off, 32);
    const float inv = 1.0f / sum;

    float* arow = attn + ((size_t)b * S_LEN + q0 + row) * S_LEN;
    for (int i = lane; i < S_LEN; i += 32) {
      const float w = srow[i] * inv;
      srow[i] = w;                            // keep normalized P in LDS for PV
      arow[i] = w;                            // final attn_weights to HBM (once)
    }
  }
  __syncthreads();

  // ===================== Phase 4: out = P @ V (one 16-col tile per wave) =========
  {
    const int d0 = wave * 16;                 // 8 waves x 16 = 128 head dims
    v8f acc = {};
    #pragma unroll 8
    for (int k0 = 0; k0 < S_LEN; k0 += 4) {
      v2f a;  a.x  = s_scores[m * S_STRIDE + k0 + koff];
              a.y  = s_scores[m * S_STRIDE + k0 + koff + 1];
      const float* vptr = Vb + (size_t)(k0 + koff) * D_HEAD + d0 + m;
      v2f bb; bb.x = vptr[0];
              bb.y = vptr[D_HEAD];
      acc = __builtin_amdgcn_wmma_f32_16x16x4_f32(
          false, a, false, bb, (short)0, acc, false, false);
    }
    float* orow = out + ((size_t)b * S_LEN + q0) * D_HEAD;
    #pragma unroll
    for (int r = 0; r < 8; ++r)
      orow[(r + row_hi) * D_HEAD + d0 + m] = acc[r];
  }
}

extern "C" void kernel_launch(void* const* d_in, const int* in_sizes, int n_in,
                              void* d_out, int out_size, void* d_ws, size_t ws_size,
                              hipStream_t stream) {
  (void)in_sizes; (void)n_in; (void)out_size; (void)d_ws; (void)ws_size;
  const float* Q = (const float*)d_in[0];
  const float* K = (const float*)d_in[1];
  const float* V = (const float*)d_in[2];
  const float* T = (const float*)d_in[3];

  float* out  = (float*)d_out;                                   // [8,4096,128]
  float* attn = out + (size_t)8 * S_LEN * D_HEAD;                // [8,4096,4096]

  const size_t shmem = (size_t)QT * S_STRIDE * sizeof(float)     // score strip
                     + 256 * sizeof(unsigned)                    // histogram
                     + QT * sizeof(float);                       // kth values
  hipFuncSetAttribute((const void*)sparse_attn_kernel,
                      hipFuncAttributeMaxDynamicSharedMemorySize, (int)shmem);

  dim3 grid(S_LEN / QT, 8);
  sparse_attn_kernel<<<grid, BLOCK, shmem, stream>>>(Q, K, V, T, out, attn);
}